// OctonionLinear_13589276525066
// MI455X (gfx1250) — compile-verified
//
#include <hip/hip_runtime.h>

// Octonion linear layer == dense fp32 GEMM:
//   y[b, i*8+ko] = sum_{j,p} x[b, j*8+p] * W[i,j,p,ko] + bias[i,ko]
//   M=512 (batch), K=4096 (=IN_OCT*8), N=4096 (=OUT_OCT*8)
// Implemented with CDNA5 V_WMMA_F32_16X16X4_F32 (full fp32 precision).

typedef __attribute__((ext_vector_type(2))) float v2f;
typedef __attribute__((ext_vector_type(8))) float v8f;

#define M_DIM 512
#define K_DIM 4096
#define N_DIM 4096
#define IN_OCT 512

#define MT 64          // workgroup M tile
#define NT 128         // workgroup N tile
#define KBLK 32        // K block staged in LDS
#define LDSS 36        // LDS row stride in floats (conflict-free: 36*r distinct mod 64)

__global__ __launch_bounds__(256, 2)
void OctonionLinear_gemm_wmma_f32(const float* __restrict__ X,
                                  const float* __restrict__ W,
                                  const float* __restrict__ Bias,
                                  float* __restrict__ Y) {
    // A tile stored [m_loc][k_loc], B tile stored [n_loc][k_loc]; both k-major.
    __shared__ float As[MT * LDSS];   // 64*36*4  = 9216 B
    __shared__ float Bs[NT * LDSS];   // 128*36*4 = 18432 B

    const int tid  = threadIdx.x;
    const int lane = tid & 31;
    const int wave = tid >> 5;      // 0..7
    const int wm   = wave >> 2;     // 0..1 : M position of wave
    const int wn   = wave & 3;      // 0..3 : N position of wave
    const int lo   = lane & 15;
    const int hi   = lane >> 4;     // 0/1 : which K half of the fragment

    const int m0 = blockIdx.y * MT;
    const int n0 = blockIdx.x * NT;

    v8f acc[2][2] = {};

    for (int kb = 0; kb < K_DIM; kb += KBLK) {
        // ---- stage A tile: MT x KBLK (rows of x are contiguous in K) ----
#pragma unroll
        for (int it = 0; it < 2; ++it) {
            const int u    = tid + it * 256;      // 0..511
            const int col4 = (u & 7) * 4;         // 0..28
            const int row  = u >> 3;              // 0..63
            const float4 v = *(const float4*)(X + (size_t)(m0 + row) * K_DIM + kb + col4);
            float* dst = &As[row * LDSS + col4];
            // stride 144 B is a multiple of 16 B; written as two b64 stores
            ((float2*)dst)[0] = make_float2(v.x, v.y);
            ((float2*)dst)[1] = make_float2(v.z, v.w);
        }
        // ---- stage B tile: KBLK x NT from W[i][j][p][ko] (contiguous 8x8 blocks) ----
#pragma unroll
        for (int it = 0; it < 4; ++it) {
            const int u   = tid + it * 256;       // 0..1023
            const int ko4 = (u & 1) * 4;          // 0 or 4
            const int p   = (u >> 1) & 7;
            const int jb  = (u >> 4) & 3;         // octonion within K block
            const int ib  = u >> 6;               // 0..15 : octonion within N tile
            const float4 v = *(const float4*)(W +
                ((size_t)(n0 / 8 + ib) * IN_OCT + (kb / 8 + jb)) * 64 + p * 8 + ko4);
            const int nloc = ib * 8 + ko4;
            const int kloc = jb * 8 + p;
            Bs[(nloc + 0) * LDSS + kloc] = v.x;
            Bs[(nloc + 1) * LDSS + kloc] = v.y;
            Bs[(nloc + 2) * LDSS + kloc] = v.z;
            Bs[(nloc + 3) * LDSS + kloc] = v.w;
        }
        __syncthreads();

        // ---- compute: 8 k-steps of K=4, 2x2 fragments each ----
#pragma unroll
        for (int ks = 0; ks < KBLK; ks += 4) {
            const int kbase = ks + hi * 2;        // even -> 8B aligned ds_load_b64
            v2f a[2], b[2];
#pragma unroll
            for (int mf = 0; mf < 2; ++mf) {
                const int row = wm * 32 + mf * 16 + lo;
                a[mf] = *(const v2f*)&As[row * LDSS + kbase];
            }
#pragma unroll
            for (int nf = 0; nf < 2; ++nf) {
                const int nrow = wn * 32 + nf * 16 + lo;
                b[nf] = *(const v2f*)&Bs[nrow * LDSS + kbase];
            }
#pragma unroll
            for (int mf = 0; mf < 2; ++mf)
#pragma unroll
                for (int nf = 0; nf < 2; ++nf)
                    acc[mf][nf] = __builtin_amdgcn_wmma_f32_16x16x4_f32(
                        /*neg_a=*/false, a[mf], /*neg_b=*/false, b[nf],
                        /*c_mod=*/(short)0, acc[mf][nf],
                        /*reuse_a=*/false, /*reuse_b=*/false);
        }
        __syncthreads();
    }

    // ---- epilogue: D layout (ISA 7.12.2): VGPR v -> M = hi*8+v, N = lo ----
#pragma unroll
    for (int nf = 0; nf < 2; ++nf) {
        const int col = n0 + wn * 32 + nf * 16 + lo;
        const float bv = Bias[col];
#pragma unroll
        for (int mf = 0; mf < 2; ++mf) {
            const int rbase = m0 + wm * 32 + mf * 16 + hi * 8;
#pragma unroll
            for (int v = 0; v < 8; ++v)
                Y[(size_t)(rbase + v) * N_DIM + col] = acc[mf][nf][v] + bv;
        }
    }
}

extern "C" void kernel_launch(void* const* d_in, const int* in_sizes, int n_in,
                              void* d_out, int out_size, void* d_ws, size_t ws_size,
                              hipStream_t stream) {
    const float* x    = (const float*)d_in[0];
    const float* w    = (const float*)d_in[1];
    const float* bias = (const float*)d_in[2];
    float* y          = (float*)d_out;

    dim3 grid(N_DIM / NT, M_DIM / MT);   // (32, 8) = 256 workgroups
    OctonionLinear_gemm_wmma_f32<<<grid, 256, 0, stream>>>(x, w, bias, y);
}